// RootsToPolynomial_67130338836724
// MI455X (gfx1250) — compile-verified
//
#include <hip/hip_runtime.h>
#include <hip/hip_bf16.h>
#include <stdint.h>

// Roots -> monic polynomial coefficients.
//   B = 262144 rows, M = 64 roots/row, output 65 coeffs/row (f32).
// One thread per row; 256 rows per block.
//   load : TDM tensor_load_to_lds with 1-DWORD row padding (stride 65 floats,
//          bank-conflict-free per-lane row reads).
//   math : fully-unrolled triangular scan, auto-packed to v_pk_fma_f32.
//   store: LDS tile (stride 65 == output row length) is the contiguous output
//          image -> one 1D TDM tensor_store_from_lds per block.

#define ROWS_PER_BLOCK 256
#define NROOTS 64
#define NCOEF 65
#define LDS_STRIDE 65               // 64 data + 1 pad -> (lane + k) % 64 banks
#define OUT_DWORDS (ROWS_PER_BLOCK * NCOEF)   // 16640, fits tile_dim0 (16-bit)

typedef uint32_t u32x4 __attribute__((ext_vector_type(4)));
typedef int      i32x8 __attribute__((ext_vector_type(8)));
typedef int      i32x4 __attribute__((ext_vector_type(4)));

#if defined(__gfx1250__) && __has_builtin(__builtin_amdgcn_tensor_load_to_lds) && \
    __has_builtin(__builtin_amdgcn_tensor_store_from_lds)
#define USE_TDM 1
#else
#define USE_TDM 0
#endif

__device__ __forceinline__ u32x4 tdm_group0(uint64_t gaddr, uint32_t lds_addr)
{
    u32x4 g0;
    g0.x = 1u;                                    // count=1 (valid user desc)
    g0.y = lds_addr;                              // LDS byte address
    g0.z = (uint32_t)gaddr;                       // global_addr[31:0]
    g0.w = ((uint32_t)(gaddr >> 32) & 0x01FFFFFFu)// global_addr[56:32]
         | (2u << 30);                            // type = 2 ("image")
    return g0;
}

__global__ __launch_bounds__(ROWS_PER_BLOCK)
void RootsToPolynomial_kernel(const float* __restrict__ x, float* __restrict__ out)
{
    __shared__ float lds[ROWS_PER_BLOCK * LDS_STRIDE];   // 66,560 B
    const int tid = threadIdx.x;
    const float* gx   = x   + (size_t)blockIdx.x * ROWS_PER_BLOCK * NROOTS;
    float*       gout = out + (size_t)blockIdx.x * (size_t)OUT_DWORDS;

#if USE_TDM
    const i32x4 gz4 = (i32x4){0, 0, 0, 0};             // groups 2/3 unused (<=2D)
    const i32x8 gz8 = (i32x8){0, 0, 0, 0, 0, 0, 0, 0}; // extra group (clang-23 form)

    // --- TDM load: 256x64 f32 tile -> LDS, pad 1 DWORD after every 64 DWORDs.
    if (tid == 0) {
        u32x4 g0 = tdm_group0((uint64_t)(uintptr_t)gx, (uint32_t)(uintptr_t)&lds[0]);
        i32x8 g1;
        g1[0] = (2 << 16)            // data_size = 4 bytes
              | (1 << 20)            // pad_enable
              | (5 << 22)            // pad_interval code 5 -> every 64 DWORDs
              | (0 << 25);           // pad_amount code 0  -> 1 DWORD
        g1[1] = (int)(64u  << 16);   // tensor_dim0 = 64   (bits 63:48)
        g1[2] = (int)(256u << 16);   // tensor_dim1 = 256  (bits 95:80)
        g1[3] = (int)(64u  << 16);   // tile_dim0   = 64   (bits 127:112)
        g1[4] = 256;                 // tile_dim1   = 256  (tile_dim2 = 0)
        g1[5] = 64;                  // tensor_dim0_stride = 64 elements
        g1[6] = 0;
        g1[7] = 0;
        __builtin_amdgcn_tensor_load_to_lds(g0, g1, gz4, gz4, gz8, 0);
        __builtin_amdgcn_s_wait_tensorcnt(0);
    }
#else
    for (int i = tid; i < ROWS_PER_BLOCK * NROOTS; i += ROWS_PER_BLOCK) {
        int r = i >> 6, c = i & 63;
        lds[r * LDS_STRIDE + c] = gx[i];
    }
#endif
    __syncthreads();

    // --- Triangular scan: a[0..k+1] are the only live coefficients at step k.
    float a[NCOEF];
    a[0] = 1.0f;
#pragma unroll
    for (int j = 1; j < NCOEF; ++j) a[j] = 0.0f;

    const float* myrow = &lds[tid * LDS_STRIDE];
#pragma unroll
    for (int k = 0; k < NROOTS; ++k) {
        float r = myrow[k];
#pragma unroll
        for (int j = k + 1; j >= 1; --j)
            a[j] = __builtin_fmaf(-r, a[j - 1], a[j]);   // a[j] -= r*a[j-1]
    }

    // --- Coefficients into own (private) LDS row; stride 65 == output row
    // length, so the LDS tile is exactly the contiguous output image.
#pragma unroll
    for (int j = 0; j < NCOEF; ++j) lds[tid * LDS_STRIDE + j] = a[j];

    __syncthreads();   // all rows written before the DMA store reads LDS

#if USE_TDM
    // --- TDM store: one 1D contiguous 16,640-DWORD LDS -> global copy.
    if (tid == 0) {
        u32x4 g0 = tdm_group0((uint64_t)(uintptr_t)gout, (uint32_t)(uintptr_t)&lds[0]);
        i32x8 g1;
        g1[0] = (2 << 16);                        // data_size = 4 B, no padding
        g1[1] = (int)((OUT_DWORDS & 0xFFFF) << 16);   // tensor_dim0 low16
        g1[2] = (int)((OUT_DWORDS >> 16) & 0xFFFF)    // tensor_dim0 high16
              | (int)(1u << 16);                      // tensor_dim1 = 1
        g1[3] = (int)((OUT_DWORDS & 0xFFFF) << 16);   // tile_dim0 = 16640
        g1[4] = 0;                                    // tile_dim1/2 unused
        g1[5] = OUT_DWORDS;                           // tensor_dim0_stride
        g1[6] = 0;
        g1[7] = 0;
        __builtin_amdgcn_tensor_store_from_lds(g0, g1, gz4, gz4, gz8, 0);
        __builtin_amdgcn_s_wait_tensorcnt(0);   // LDS must outlive the DMA
    }
#else
#pragma unroll
    for (int j = 0; j < NCOEF; ++j) {
        int idx = j * ROWS_PER_BLOCK + tid;
        __builtin_nontemporal_store(lds[idx], &gout[idx]);
    }
#endif
}

extern "C" void kernel_launch(void* const* d_in, const int* in_sizes, int n_in,
                              void* d_out, int out_size, void* d_ws, size_t ws_size,
                              hipStream_t stream)
{
    (void)n_in; (void)out_size; (void)d_ws; (void)ws_size;
    const float* x = (const float*)d_in[0];
    float* out = (float*)d_out;

    int n_elems = in_sizes[0];               // B * 64
    int rows = n_elems / NROOTS;             // 262144
    int grid = rows / ROWS_PER_BLOCK;        // 1024

    RootsToPolynomial_kernel<<<dim3(grid), dim3(ROWS_PER_BLOCK), 0, stream>>>(x, out);
}